// GRU2D_24653112279539
// MI455X (gfx1250) — compile-verified
//
#include <hip/hip_runtime.h>
#include <math.h>

#define TSIZE 32
#define HALO 3
#define LW 38            // TSIZE + 2*HALO
#define LSTR 40          // padded LDS row stride (floats)
#define IMG_H 1024
#define IMG_W 1024
#define BATCH 8
#define NTOT (BATCH * IMG_H * IMG_W)
#define TWO_OVER_PI 0.63661977236758134308f

typedef float v2f __attribute__((ext_vector_type(2)));
typedef float v8f __attribute__((ext_vector_type(8)));

#if __has_builtin(__builtin_amdgcn_wmma_f32_16x16x4_f32)
#define USE_WMMA 1
#else
#define USE_WMMA 0
#endif

#if __has_builtin(__builtin_amdgcn_global_load_async_to_lds_b32)
#define USE_ASYNC 1
typedef __attribute__((address_space(1))) int as_g_i32;
typedef __attribute__((address_space(3))) int as_l_i32;
#else
#define USE_ASYNC 0
#endif

// Fixed curvature stencils, tap index t = ky*3 + kx (cross-correlation, SAME/zero pad)
__device__ const float c_stencil[5][9] = {
    {-1.f, 0.f, 1.f, -1.f, 0.f, 1.f, -1.f, 0.f, 1.f},   // SX
    {-1.f, -1.f, -1.f, 0.f, 0.f, 0.f, 1.f, 1.f, 1.f},   // SY
    {0.f, 0.f, 0.f, 1.f, -2.f, 1.f, 0.f, 0.f, 0.f},     // DXX
    {0.f, 1.f, 0.f, 0.f, -2.f, 0.f, 0.f, -1.f, 0.f},    // DYY
    {0.f, -1.f, 1.f, 0.f, 1.f, -1.f, 0.f, 0.f, 0.f}     // DXY
};

__device__ __forceinline__ void wait_async_lds() {
#if USE_ASYNC
#if __has_builtin(__builtin_amdgcn_s_wait_asynccnt)
    __builtin_amdgcn_s_wait_asynccnt(0);
#else
    asm volatile("s_wait_asynccnt 0" ::: "memory");
#endif
#endif
}

__device__ __forceinline__ float sigmoidf_(float x) {
    return 1.f / (1.f + expf(-x));
}

// -------------------- global reduction: sum(H), sum(img*H), sum(img) ----------------
__global__ __launch_bounds__(256) void ls_reduce(const float* __restrict__ phi,
                                                 const float* __restrict__ img,
                                                 float* __restrict__ acc) {
    __shared__ float rbuf[3][256];
    const int tid = threadIdx.x;
    float h = 0.f, ih = 0.f, is = 0.f;
    for (int i = blockIdx.x * blockDim.x + tid; i < NTOT; i += gridDim.x * blockDim.x) {
        float p = phi[i];
        float im = img[i];
        float Hv = 0.5f * (1.0f + TWO_OVER_PI * atanf(p * 32.0f));  // /E_LS == *32
        h += Hv;
        ih += im * Hv;
        is += im;
    }
    rbuf[0][tid] = h; rbuf[1][tid] = ih; rbuf[2][tid] = is;
    __syncthreads();
    for (int s = 128; s > 0; s >>= 1) {
        if (tid < s) {
            rbuf[0][tid] += rbuf[0][tid + s];
            rbuf[1][tid] += rbuf[1][tid + s];
            rbuf[2][tid] += rbuf[2][tid + s];
        }
        __syncthreads();
    }
    if (tid == 0) {
        atomicAdd(&acc[0], rbuf[0][0]);
        atomicAdd(&acc[1], rbuf[1][0]);
        atomicAdd(&acc[2], rbuf[2][0]);
    }
}

// -------------------- fully fused per-iteration update ------------------------------
__global__ __launch_bounds__(256) void ls_update(
    const float* __restrict__ phi_in, const float* __restrict__ img,
    const float* __restrict__ wz_u, const float* __restrict__ wz_w,
    const float* __restrict__ wr_u, const float* __restrict__ wr_w,
    const float* __restrict__ wh_u, const float* __restrict__ wh_w,
    const float* __restrict__ wg_u, const float* __restrict__ wg_w,
    const float* __restrict__ bz_u, const float* __restrict__ bz_w,
    const float* __restrict__ br_u, const float* __restrict__ br_w,
    const float* __restrict__ bh_u, const float* __restrict__ bh_w,
    const float* __restrict__ acc, float* __restrict__ phi_out)
{
    __shared__ float sPhi[LW * LSTR];     // phi tile + halo3
    __shared__ float sImg[LW * LSTR];     // img tile + halo3
    __shared__ float sInp[36 * LSTR];     // inp on tile+2 halo
    __shared__ float sRphi[34 * LSTR];    // r*phi on tile+1 halo
    __shared__ float sZ[32 * 33];         // z gate on tile
    __shared__ float sHu[32 * 33];        // conv(inp, wh_u) on tile
    __shared__ float wB1[12 * 16];        // stencil bank B matrix [K=12][N=16], ch 0..4 used
    __shared__ float wB2[12 * 16];        // gate bank   B matrix, ch 0=z 1=r 2=h
    __shared__ float sWhw[9], sWgu[9], sWgw[9];
    __shared__ float sBz, sBr, sBh, sC1, sC2;
#if USE_WMMA
    __shared__ float stage[8 * 16 * 17];  // per-wave D-matrix staging
#endif

    const int tid = threadIdx.x;
    const int x0 = blockIdx.x * TSIZE;
    const int y0 = blockIdx.y * TSIZE;
    const int base = blockIdx.z * (IMG_H * IMG_W);

    // ---- weight prep (cheap, overlapped with tile load before the barrier)
    for (int i = tid; i < 192; i += 256) {
        int k = i >> 4, n = i & 15;
        float w1 = 0.f, w2 = 0.f;
        if (k < 9) {
            if (n < 5) w1 = c_stencil[n][k];
            if (n == 0) w2 = wz_u[k] + wz_w[k];        // conv(x,wu)+conv(x,ww)=conv(x,wu+ww)
            else if (n == 1) w2 = wr_u[k] + wr_w[k];
            else if (n == 2) w2 = wh_u[k];
        }
        wB1[i] = w1; wB2[i] = w2;
    }
    if (tid < 9) { sWhw[tid] = wh_w[tid]; sWgu[tid] = wg_u[tid]; sWgw[tid] = wg_w[tid]; }
    if (tid == 0) {
        sBz = bz_u[0] + bz_w[0];
        sBr = br_u[0] + br_w[0];
        sBh = bh_u[0] + bh_w[0];
        float sH = acc[0], sIH = acc[1], sI = acc[2];
        sC1 = sIH / sH;
        sC2 = (sI - sIH) / ((float)NTOT - sH);
    }

    // ---- tile load with halo + zero pad (async DMA into LDS where available)
    for (int i = tid; i < LW * LW; i += 256) {
        int r = i / LW, c = i - r * LW;
        int gy = y0 - HALO + r, gx = x0 - HALO + c;
        int di = r * LSTR + c;
        if (gy >= 0 && gy < IMG_H && gx >= 0 && gx < IMG_W) {
            int gi = base + gy * IMG_W + gx;
#if USE_ASYNC
            __builtin_amdgcn_global_load_async_to_lds_b32(
                (as_g_i32*)&phi_in[gi], (as_l_i32*)&sPhi[di], 0, 0);
            __builtin_amdgcn_global_load_async_to_lds_b32(
                (as_g_i32*)&img[gi], (as_l_i32*)&sImg[di], 0, 0);
#else
            sPhi[di] = phi_in[gi];
            sImg[di] = img[gi];
#endif
        } else {
            sPhi[di] = 0.f;
            sImg[di] = 0.f;
        }
    }
    wait_async_lds();
    __syncthreads();

    const float c1 = sC1, c2 = sC2;
    const int wave = tid >> 5;
    const int lane = tid & 31;
    const int m16 = lane & 15;
    const int khalf = (lane >> 4) << 1;   // lanes 0-15 -> K 0/1, lanes 16-31 -> K 2/3

#if USE_WMMA
    float* stg = &stage[wave * (16 * 17)];

    // Hoist loop-invariant B-matrix operands (per-lane weight pairs) into registers:
    // only depend on (ch, khalf, m16), not on the group index.
    v2f b1v[3], b2v[3];
#pragma unroll
    for (int ch = 0; ch < 3; ++ch) {
        int t0 = ch * 4 + khalf, t1 = t0 + 1;
        b1v[ch].x = wB1[t0 * 16 + m16];
        b1v[ch].y = wB1[t1 * 16 + m16];
        b2v[ch].x = wB2[t0 * 16 + m16];
        b2v[ch].y = wB2[t1 * 16 + m16];
    }

    // ================= stage 1: inp over 36x36 (5 curvature stencils via WMMA) =======
    // 1296 points = 81 groups of 16; 8 waves, 11 uniform iterations.
    for (int it = 0; it < 11; ++it) {
        int g = it * 8 + wave;
        int gc = g < 81 ? g : 80;
        int p = gc * 16 + m16;
        int rr = p / 36, cc = p - rr * 36;    // region coords -> sPhi coord +1
        int pr = rr + 1, pc = cc + 1;
        v8f d = {0.f, 0.f, 0.f, 0.f, 0.f, 0.f, 0.f, 0.f};
#pragma unroll
        for (int ch = 0; ch < 3; ++ch) {
            int t0 = ch * 4 + khalf, t1 = t0 + 1;
            int ta = t0 < 9 ? t0 : 0, tb = t1 < 9 ? t1 : 0;
            v2f a;
            float va = sPhi[(pr + ta / 3 - 1) * LSTR + (pc + ta % 3 - 1)];
            float vb = sPhi[(pr + tb / 3 - 1) * LSTR + (pc + tb % 3 - 1)];
            a.x = t0 < 9 ? va : 0.f;
            a.y = t1 < 9 ? vb : 0.f;
            d = __builtin_amdgcn_wmma_f32_16x16x4_f32(false, a, false, b1v[ch], (short)0, d,
                                                      false, false);
        }
#pragma unroll
        for (int j = 0; j < 8; ++j) {
            int mr = (lane < 16) ? j : (j + 8);
            stg[mr * 17 + m16] = d[j];
        }
        asm volatile("s_wait_dscnt 0" ::: "memory");
        if (lane < 16 && g < 81) {
            float gxv = stg[m16 * 17 + 0], gyv = stg[m16 * 17 + 1];
            float dxx = stg[m16 * 17 + 2], dyy = stg[m16 * 17 + 3], dxy = stg[m16 * 17 + 4];
            float it4 = gxv * gxv + gyv * gyv;
            float denom = it4 * sqrtf(it4) + (it4 == 0.f ? 1.f : 0.f);
            float curv = (dxx * gyv * gyv + 2.f * gxv * gyv * dxy + dyy * gxv * gxv) / denom;
            float cA = 0.f, cB = 0.f;
#pragma unroll
            for (int t = 0; t < 9; ++t) {
                float iv = sImg[(pr + t / 3 - 1) * LSTR + (pc + t % 3 - 1)];
                float d1 = iv - c1, d2 = iv - c2;
                cA += sWgu[t] * d1 * d1;
                cB += sWgw[t] * d2 * d2;
            }
            sInp[rr * LSTR + cc] = curv - cA + cB;
        }
    }
    __syncthreads();

    // ================= stage 2: z / r / h-pre over 34x34 (gate bank via WMMA) ========
    // 1156 points -> 73 groups (last partial); 10 uniform iterations.
    for (int it = 0; it < 10; ++it) {
        int g = it * 8 + wave;
        int gc = g < 73 ? g : 72;
        int p0 = gc * 16 + m16;
        int p = p0 < 1156 ? p0 : 1155;
        int r2 = p / 34, c2 = p - r2 * 34;    // region2 coords: sPhi +2, sInp +1
        int ir = r2 + 1, ic = c2 + 1;
        v8f d = {0.f, 0.f, 0.f, 0.f, 0.f, 0.f, 0.f, 0.f};
#pragma unroll
        for (int ch = 0; ch < 3; ++ch) {
            int t0 = ch * 4 + khalf, t1 = t0 + 1;
            int ta = t0 < 9 ? t0 : 0, tb = t1 < 9 ? t1 : 0;
            v2f a;
            float va = sInp[(ir + ta / 3 - 1) * LSTR + (ic + ta % 3 - 1)];
            float vb = sInp[(ir + tb / 3 - 1) * LSTR + (ic + tb % 3 - 1)];
            a.x = t0 < 9 ? va : 0.f;
            a.y = t1 < 9 ? vb : 0.f;
            d = __builtin_amdgcn_wmma_f32_16x16x4_f32(false, a, false, b2v[ch], (short)0, d,
                                                      false, false);
        }
#pragma unroll
        for (int j = 0; j < 8; ++j) {
            int mr = (lane < 16) ? j : (j + 8);
            stg[mr * 17 + m16] = d[j];
        }
        asm volatile("s_wait_dscnt 0" ::: "memory");
        if (lane < 16 && g < 73 && p0 < 1156) {
            float zpre = stg[m16 * 17 + 0];
            float rpre = stg[m16 * 17 + 1];
            float hpre = stg[m16 * 17 + 2];
            float rv = sigmoidf_(rpre + sBr);
            sRphi[r2 * LSTR + c2] = rv * sPhi[(r2 + 2) * LSTR + (c2 + 2)];
            if (r2 >= 1 && r2 < 33 && c2 >= 1 && c2 < 33) {
                int ty = r2 - 1, tx = c2 - 1;
                sZ[ty * 33 + tx] = sigmoidf_(zpre + sBz);
                sHu[ty * 33 + tx] = hpre;
            }
        }
    }
    __syncthreads();

#else  // ----------------------------- scalar fallback (no WMMA builtin) -------------
    for (int p = tid; p < 36 * 36; p += 256) {
        int rr = p / 36, cc = p - rr * 36;
        int pr = rr + 1, pc = cc + 1;
        float ch[5] = {0.f, 0.f, 0.f, 0.f, 0.f};
#pragma unroll
        for (int t = 0; t < 9; ++t) {
            float v = sPhi[(pr + t / 3 - 1) * LSTR + (pc + t % 3 - 1)];
#pragma unroll
            for (int n = 0; n < 5; ++n) ch[n] += wB1[t * 16 + n] * v;
        }
        float gxv = ch[0], gyv = ch[1], dxx = ch[2], dyy = ch[3], dxy = ch[4];
        float it4 = gxv * gxv + gyv * gyv;
        float denom = it4 * sqrtf(it4) + (it4 == 0.f ? 1.f : 0.f);
        float curv = (dxx * gyv * gyv + 2.f * gxv * gyv * dxy + dyy * gxv * gxv) / denom;
        float cA = 0.f, cB = 0.f;
#pragma unroll
        for (int t = 0; t < 9; ++t) {
            float iv = sImg[(pr + t / 3 - 1) * LSTR + (pc + t % 3 - 1)];
            float d1 = iv - c1, d2 = iv - c2;
            cA += sWgu[t] * d1 * d1;
            cB += sWgw[t] * d2 * d2;
        }
        sInp[rr * LSTR + cc] = curv - cA + cB;
    }
    __syncthreads();

    for (int p = tid; p < 34 * 34; p += 256) {
        int r2 = p / 34, c2 = p - r2 * 34;
        int ir = r2 + 1, ic = c2 + 1;
        float zp = 0.f, rp = 0.f, hp = 0.f;
#pragma unroll
        for (int t = 0; t < 9; ++t) {
            float v = sInp[(ir + t / 3 - 1) * LSTR + (ic + t % 3 - 1)];
            zp += wB2[t * 16 + 0] * v;
            rp += wB2[t * 16 + 1] * v;
            hp += wB2[t * 16 + 2] * v;
        }
        float rv = sigmoidf_(rp + sBr);
        sRphi[r2 * LSTR + c2] = rv * sPhi[(r2 + 2) * LSTR + (c2 + 2)];
        if (r2 >= 1 && r2 < 33 && c2 >= 1 && c2 < 33) {
            int ty = r2 - 1, tx = c2 - 1;
            sZ[ty * 33 + tx] = sigmoidf_(zp + sBz);
            sHu[ty * 33 + tx] = hp;
        }
    }
    __syncthreads();
#endif

    // ================= stage 3: ht = tanh(hu + bh + conv(r*phi, wh_w)); blend ========
    for (int idx = tid; idx < TSIZE * TSIZE; idx += 256) {
        int ty = idx >> 5, tx = idx & 31;
        float hw = 0.f;
#pragma unroll
        for (int t = 0; t < 9; ++t)
            hw += sWhw[t] * sRphi[(ty + t / 3) * LSTR + (tx + t % 3)];
        float ht = tanhf(sHu[ty * 33 + tx] + sBh + hw);
        float z = sZ[ty * 33 + tx];
        float ph = sPhi[(ty + 3) * LSTR + (tx + 3)];
        phi_out[base + (y0 + ty) * IMG_W + (x0 + tx)] = (1.f - z) * ht + z * ph;
    }
}

// -------------------- host-side launch sequence -------------------------------------
extern "C" void kernel_launch(void* const* d_in, const int* in_sizes, int n_in,
                              void* d_out, int out_size, void* d_ws, size_t ws_size,
                              hipStream_t stream) {
    (void)in_sizes; (void)n_in; (void)out_size; (void)ws_size;
    const float* phi0 = (const float*)d_in[0];
    const float* img  = (const float*)d_in[1];
    const float* w[8];
    for (int i = 0; i < 8; ++i) w[i] = (const float*)d_in[2 + i];
    const float* b[6];
    for (int i = 0; i < 6; ++i) b[i] = (const float*)d_in[10 + i];
    float* out = (float*)d_out;
    float* acc = (float*)d_ws;                       // 4 floats
    float* tmp = (float*)((char*)d_ws + 256);        // 32 MB ping buffer

    const float* cur = phi0;
    float* dsts[5] = {out, tmp, out, tmp, out};      // never in==out; final lands in d_out
    dim3 ugrid(IMG_W / TSIZE, IMG_H / TSIZE, BATCH);
    for (int it = 0; it < 5; ++it) {
        (void)hipMemsetAsync(acc, 0, 4 * sizeof(float), stream);
        ls_reduce<<<2048, 256, 0, stream>>>(cur, img, acc);
        ls_update<<<ugrid, 256, 0, stream>>>(cur, img,
            w[0], w[1], w[2], w[3], w[4], w[5], w[6], w[7],
            b[0], b[1], b[2], b[3], b[4], b[5], acc, dsts[it]);
        cur = dsts[it];
    }
}